// Chf_Likelihood_Loss_20478404067560
// MI455X (gfx1250) — compile-verified
//
#include <hip/hip_runtime.h>

typedef __attribute__((ext_vector_type(2))) float v2f;
typedef __attribute__((ext_vector_type(8))) float v8f;

#define LDW 65            // LDS row stride (64 + 1 pad -> conflict-free strided loads)
#define S2 60             // 2*CHF_STEP
#define HW 64             // H == W == 64
#define NPIX (HW * HW)    // 4096

__device__ __forceinline__ v8f wmma_f32(v2f a, v2f b, v8f c) {
  // V_WMMA_F32_16X16X4_F32 : D = A(16x4) * B(4x16) + C(16x16), all f32
  return __builtin_amdgcn_wmma_f32_16x16x4_f32(
      /*neg_a=*/false, a, /*neg_b=*/false, b,
      /*c_mod=*/(short)0, c, /*reuse_a=*/false, /*reuse_b=*/false);
}

// One workgroup per batch element. 256 threads = 8 wave32.
__global__ void __launch_bounds__(256)
chf_loss_kernel(const float* __restrict__ dnn, const float* __restrict__ gt,
                float* __restrict__ loss) {
  __shared__ float sD[HW * LDW];   // difference map, D[w][h] = dnn[b,h,w]-gt[b,h,w]
  __shared__ float sC[HW * LDW];   // cos table, rows 60..63 zero
  __shared__ float sS[HW * LDW];   // sin table, rows 60..63 zero
  __shared__ float sPc[HW * LDW];  // stage-1 result Pc[w][i]
  __shared__ float sPs[HW * LDW];  // stage-1 result Ps[w][i]
  __shared__ float ssum;

  const int b    = blockIdx.x;
  const int tid  = threadIdx.x;
  const int wid  = tid >> 5;       // wave id 0..7
  const int lane = tid & 31;
  const int m     = lane & 15;               // row/col within 16-tile
  const int khalf = (lane >> 4) << 1;        // 0 for lanes 0-15, 2 for lanes 16-31

  if (tid == 0) ssum = 0.0f;

  // ---- Stage 0: trig tables + difference map into LDS ----
  const float* pout = dnn + (size_t)b * NPIX;
  const float* pgt  = gt  + (size_t)b * NPIX;
  for (int idx = tid; idx < NPIX; idx += 256) {
    int r = idx >> 6;   // freq index (tables) / h (map)
    int c = idx & 63;   // h (tables) / w (map)
    // tables: C[r][c] = cos(f_r * (c+0.5)), zero-padded past 60 rows
    float f   = (float)(r - 30) * 0.01f;
    float ang = f * ((float)c + 0.5f);
    bool v = (r < S2);
    sC[r * LDW + c] = v ? __cosf(ang) : 0.0f;
    sS[r * LDW + c] = v ? __sinf(ang) : 0.0f;
    // difference map: coalesced global read (c = w contiguous), strided LDS write
    sD[c * LDW + r] = pout[idx] - pgt[idx];   // sD[w][h] = dnn[b,h,w]-gt[b,h,w]
  }
  __syncthreads();

  // ---- Stage 1: Pc[w,i] = sum_h D[w,h]*C[i,h] ; Ps with S.  (GEMM, B = table^T)
  // 2 matrices * 16 tiles = 32 jobs over 8 waves
  for (int t = 0; t < 4; ++t) {
    int job = t * 8 + wid;
    int matsel = job >> 4;           // 0 -> Pc, 1 -> Ps
    int tile = job & 15;
    int mt = tile >> 2, nt = tile & 3;
    const float* Btab = matsel ? sS : sC;
    v8f acc = {};
    for (int k0 = 0; k0 < HW; k0 += 4) {
      v2f a, bf;
      int ar = (mt * 16 + m) * LDW + k0 + khalf;
      a.x = sD[ar];  a.y = sD[ar + 1];
      int br = (nt * 16 + m) * LDW + k0 + khalf;   // B[k][n] = tab[n][k]
      bf.x = Btab[br]; bf.y = Btab[br + 1];
      acc = wmma_f32(a, bf, acc);
    }
    float* P = matsel ? sPs : sPc;
    int rbase = mt * 16 + ((lane >> 4) << 3);      // M = v (+8 for upper half-wave)
    int col   = nt * 16 + m;
    #pragma unroll
    for (int v2 = 0; v2 < 8; ++v2) P[(rbase + v2) * LDW + col] = acc[v2];
  }
  __syncthreads();

  // ---- Stage 2: R = C*Pc - S*Ps ; I = S*Pc + C*Ps ; accumulate sum(R^2+I^2)
  // 16 tiles over 8 waves
  float part = 0.0f;
  for (int t = 0; t < 2; ++t) {
    int tile = t * 8 + wid;
    int mt = tile >> 2, nt = tile & 3;
    v8f accR = {}, accI = {};
    for (int k0 = 0; k0 < HW; k0 += 4) {
      int ar = (mt * 16 + m) * LDW + k0 + khalf;
      v2f aC, aS, aSn, bPc, bPs;
      aC.x = sC[ar]; aC.y = sC[ar + 1];
      aS.x = sS[ar]; aS.y = sS[ar + 1];
      aSn.x = -aS.x; aSn.y = -aS.y;
      int b0 = (k0 + khalf) * LDW + nt * 16 + m;   // B[k][n] = P[k][n], row-contig
      bPc.x = sPc[b0]; bPc.y = sPc[b0 + LDW];
      bPs.x = sPs[b0]; bPs.y = sPs[b0 + LDW];
      accR = wmma_f32(aC,  bPc, accR);
      accR = wmma_f32(aSn, bPs, accR);
      accI = wmma_f32(aS,  bPc, accI);
      accI = wmma_f32(aC,  bPs, accI);
    }
    #pragma unroll
    for (int v2 = 0; v2 < 8; ++v2)
      part += accR[v2] * accR[v2] + accI[v2] * accI[v2];
  }

  // ---- Reduction: block-local LDS atomic, then one global atomic per block
  __syncthreads();               // ensure ssum init visible & stage reads done
  atomicAdd(&ssum, part);
  __syncthreads();
  if (tid == 0) atomicAdd(loss, ssum * 0.125f);   // mean over B=8, SCALE=1
}

extern "C" void kernel_launch(void* const* d_in, const int* in_sizes, int n_in,
                              void* d_out, int out_size, void* d_ws, size_t ws_size,
                              hipStream_t stream) {
  const float* dnn = (const float*)d_in[0];
  const float* gt  = (const float*)d_in[1];
  float* loss = (float*)d_out;
  hipMemsetAsync(loss, 0, sizeof(float) * (size_t)out_size, stream);
  chf_loss_kernel<<<dim3(8), dim3(256), 0, stream>>>(dnn, gt, loss);
}